// HiResPrecipNet_57174604645056
// MI455X (gfx1250) — compile-verified
//
#include <hip/hip_runtime.h>
#include <cstddef>

// ---------------------------------------------------------------------------
// HiResPrecipNet on gfx1250 (MI455X): GATv2 stack with fp32 WMMA GEMMs.
// wave32 everywhere. GEMMs use V_WMMA_F32_16X16X4_F32 (exact fp32 math).
// Inner GEMM loop is branch-free: OOB handled by clamped addresses + 0/1
// float masks so EXEC stays all-ones at every WMMA (ISA 7.12 requirement).
// ---------------------------------------------------------------------------

typedef float v2f __attribute__((ext_vector_type(2)));
typedef float v8f __attribute__((ext_vector_type(8)));

#define N_LOW  5000
#define N_HIGH 50000
#define E_LH   200000
#define E_HH   400000

// ---- ordered-uint mapping for float atomic max -----------------------------
__device__ __forceinline__ unsigned f2ord(float f) {
  unsigned u = __float_as_uint(f);
  return (u & 0x80000000u) ? ~u : (u | 0x80000000u);
}
__device__ __forceinline__ float ord2f(unsigned o) {
  unsigned u = (o & 0x80000000u) ? (o & 0x7FFFFFFFu) : ~o;
  return __uint_as_float(u);
}

__device__ __forceinline__ int imin(int a, int b) { return a < b ? a : b; }

// ---------------------------------------------------------------------------
// Generic GEMM: C[M,N] = A[M,K] @ W[K,N] + bias[N]  (optional ReLU)
// 128 threads = 4 waves per block; each wave computes one 16x16 tile with
// V_WMMA_F32_16X16X4_F32. Main loop steps K by 8 (2 WMMAs), unconditional
// loads; grid-uniform step-4 + masked-tail epilogues cover any K.
// f32 WMMA VGPR layout (ISA 7.12.2):
//   A 16x4 : lanes 0-15 rows, vgpr0/1 = K+0/K+1; lanes16-31 = K+2/K+3
//   B 4x16 : lanes hold N columns, vgpr0/1 = K rows (same half split)
//   C 16x16: vgpr r = row (tm + r + 8*half), col = tn + (lane&15)
// ---------------------------------------------------------------------------
__global__ void __launch_bounds__(128)
gemm_f32_wmma(const float* __restrict__ A, const float* __restrict__ W,
              const float* __restrict__ bias, float* __restrict__ Cout,
              int M, int K, int N, int relu)
{
  const int lane = threadIdx.x & 31;
  const int wave = threadIdx.x >> 5;
  const int tn   = (blockIdx.x * 4 + wave) * 16;
  const int tm   = blockIdx.y * 16;
  const int half = lane >> 4;
  const int l16  = lane & 15;

  const int   arow  = tm + l16;
  const int   bcol  = tn + l16;
  const float amask = (arow < M) ? 1.f : 0.f;
  const float bmask = (bcol < N) ? 1.f : 0.f;
  const float* Ar   = A + (size_t)imin(arow, M - 1) * K;   // clamped, always valid
  const float* Wc   = W + imin(bcol, N - 1);

  v8f acc = {0.f, 0.f, 0.f, 0.f, 0.f, 0.f, 0.f, 0.f};

  const int K4 = K & ~3;
  const int K8 = K & ~7;
  int k = 0;

  // Main loop: step 8, fully unconditional loads, 2 WMMAs per iteration.
  for (; k < K8; k += 8) {
    const int ak0 = k + half * 2;
    const int ak1 = ak0 + 4;
    __builtin_prefetch(Ar + imin(ak0 + 8, K - 1), 0, 3);   // global_prefetch_b8
    v2f a0, a1, b0, b1;
    a0.x = Ar[ak0];     a0.y = Ar[ak0 + 1];
    a1.x = Ar[ak1];     a1.y = Ar[ak1 + 1];
    b0.x = Wc[(size_t)ak0 * N];       b0.y = Wc[(size_t)(ak0 + 1) * N];
    b1.x = Wc[(size_t)ak1 * N];       b1.y = Wc[(size_t)(ak1 + 1) * N];
    a0 *= amask; a1 *= amask; b0 *= bmask; b1 *= bmask;
    acc = __builtin_amdgcn_wmma_f32_16x16x4_f32(false, a0, false, b0,
                                                (short)0, acc, false, false);
    acc = __builtin_amdgcn_wmma_f32_16x16x4_f32(false, a1, false, b1,
                                                (short)0, acc, false, false);
  }

  // Grid-uniform step-4 epilogue (K4 > K8 for K % 8 in [4,7]).
  if (k < K4) {
    const int ak = k + half * 2;
    v2f a, b;
    a.x = Ar[ak] * amask;             a.y = Ar[ak + 1] * amask;
    b.x = Wc[(size_t)ak * N] * bmask; b.y = Wc[(size_t)(ak + 1) * N] * bmask;
    acc = __builtin_amdgcn_wmma_f32_16x16x4_f32(false, a, false, b,
                                                (short)0, acc, false, false);
    k += 4;
  }

  // Grid-uniform masked tail (K % 4 != 0): clamped indices + per-elem masks.
  if (k < K) {
    const int ak = k + half * 2;
    const int a0i = imin(ak, K - 1);
    const int a1i = imin(ak + 1, K - 1);
    const float m0 = (ak     < K) ? amask : 0.f;
    const float m1 = (ak + 1 < K) ? amask : 0.f;
    const float n0 = (ak     < K) ? bmask : 0.f;
    const float n1 = (ak + 1 < K) ? bmask : 0.f;
    v2f a, b;
    a.x = Ar[a0i] * m0;               a.y = Ar[a1i] * m1;
    b.x = Wc[(size_t)a0i * N] * n0;   b.y = Wc[(size_t)a1i * N] * n1;
    acc = __builtin_amdgcn_wmma_f32_16x16x4_f32(false, a, false, b,
                                                (short)0, acc, false, false);
  }

  const float bv = (bcol < N) ? bias[bcol] : 0.f;
#pragma unroll
  for (int r = 0; r < 8; ++r) {
    const int row = tm + r + half * 8;
    if (row < M && bcol < N) {
      float v = acc[r] + bv;
      if (relu) v = fmaxf(v, 0.f);
      Cout[(size_t)row * N + bcol] = v;
    }
  }
}

// ---------------------------------------------------------------------------
__global__ void fill_u32(unsigned* __restrict__ p, unsigned v, int n)
{
  int i = blockIdx.x * blockDim.x + threadIdx.x;
  if (i < n) p[i] = v;
}

// One wave per edge: logit[e,h] = sum_c leakyrelu(hl[src]+hr[dst]) * att[h,c];
// then ordered-uint atomicMax into per-(dst,head) running max.
__global__ void edge_logit_max(const float* __restrict__ hl, const float* __restrict__ hr,
                               const float* __restrict__ att,
                               const int* __restrict__ si, const int* __restrict__ di,
                               int Er, int nloop, int H, int C,
                               float* __restrict__ logit, unsigned* __restrict__ m)
{
  const int gw   = (blockIdx.x * blockDim.x + threadIdx.x) >> 5;
  const int lane = threadIdx.x & 31;
  const int Et   = Er + nloop;
  if (gw >= Et) return;

  int ss, dd;
  if (gw < Er) { ss = si[gw]; dd = di[gw]; }
  else         { ss = dd = gw - Er; }

  const int D = H * C;
  const float* hs = hl + (size_t)ss * D;
  const float* hd = hr + (size_t)dd * D;

  for (int h = 0; h < H; ++h) {
    float acc = 0.f;
    for (int c = lane; c < C; c += 32) {
      float z = hs[h * C + c] + hd[h * C + c];
      z = (z > 0.f) ? z : 0.2f * z;              // leaky_relu(., 0.2)
      acc += z * att[h * C + c];
    }
#pragma unroll
    for (int off = 16; off; off >>= 1) acc += __shfl_xor(acc, off, 32);
    if (lane == 0) {
      logit[gw * H + h] = acc;
      atomicMax(&m[dd * H + h], f2ord(acc));
    }
  }
}

// ex = exp(logit - m[dst]); segment-sum into s; count edges per dst (head 0).
__global__ void edge_exp_sum(float* __restrict__ logit, const unsigned* __restrict__ m,
                             const int* __restrict__ di, int Er, int nloop, int H,
                             float* __restrict__ s, float* __restrict__ cnt)
{
  const int idx = blockIdx.x * blockDim.x + threadIdx.x;
  const int Et  = Er + nloop;
  if (idx >= Et * H) return;
  const int e = idx / H;
  const int h = idx - e * H;
  const int dd = (e < Er) ? di[e] : (e - Er);
  const float ex = expf(logit[idx] - ord2f(m[dd * H + h]));
  logit[idx] = ex;                         // overwrite logit with ex
  atomicAdd(&s[dd * H + h], ex);
  if (h == 0) atomicAdd(&cnt[dd], 1.0f);
}

// out[dst, c] += hl[src, c] * ex[e,h]/(s[dst,h]+1e-16)
__global__ void edge_scatter(const float* __restrict__ hl,
                             const int* __restrict__ si, const int* __restrict__ di,
                             int Er, int nloop, int H, int C,
                             const float* __restrict__ ex, const float* __restrict__ s,
                             float* __restrict__ out)
{
  const int idx = blockIdx.x * blockDim.x + threadIdx.x;
  const int Et  = Er + nloop;
  const int D   = H * C;
  if (idx >= Et * D) return;
  const int e = idx / D;
  const int c = idx - e * D;
  const int h = c / C;
  int ss, dd;
  if (e < Er) { ss = si[e]; dd = di[e]; }
  else        { ss = dd = e - Er; }
  const float alpha = ex[e * H + h] / (s[dd * H + h] + 1e-16f);
  atomicAdd(&out[(size_t)dd * D + c], hl[(size_t)ss * D + c] * alpha);
}

// x[n, 0] = zstd[n]; x[n, 1+j] = out[n, j]/max(cnt,1) + bias[j]   (D = 256)
__global__ void finalize_concat(const float* __restrict__ out, const float* __restrict__ cnt,
                                const float* __restrict__ bias, const float* __restrict__ zstd,
                                float* __restrict__ x, int n)
{
  const int idx = blockIdx.x * blockDim.x + threadIdx.x;
  if (idx >= n * 257) return;
  const int r = idx / 257;
  const int c = idx - r * 257;
  float v;
  if (c == 0) v = zstd[r];
  else v = out[(size_t)r * 256 + (c - 1)] / fmaxf(cnt[r], 1.0f) + bias[c - 1];
  x[idx] = v;
}

// x[n, c] = out[n, c]/max(cnt,1) + bias[c]
__global__ void finalize_mean(const float* __restrict__ out, const float* __restrict__ cnt,
                              const float* __restrict__ bias, float* __restrict__ x,
                              int n, int D)
{
  const int idx = blockIdx.x * blockDim.x + threadIdx.x;
  if (idx >= n * D) return;
  const int r = idx / D;
  const int c = idx - r * D;
  x[idx] = out[idx] / fmaxf(cnt[r], 1.0f) + bias[c];
}

// BatchNorm statistics: per-channel sum & sumsq (block partials in LDS).
__global__ void bn_reduce(const float* __restrict__ x, int n, int d,
                          float* __restrict__ sums)
{
  extern __shared__ float sh[];                // 2*d floats
  const int r0 = blockIdx.x * 128;
  for (int i = threadIdx.x; i < 2 * d; i += blockDim.x) sh[i] = 0.f;
  __syncthreads();
  for (int i = threadIdx.x; i < 128 * d; i += blockDim.x) {
    const int r = r0 + i / d;
    const int c = i - (i / d) * d;
    if (r < n) {
      const float v = x[(size_t)r * d + c];
      atomicAdd(&sh[c], v);
      atomicAdd(&sh[d + c], v * v);
    }
  }
  __syncthreads();
  for (int c = threadIdx.x; c < d; c += blockDim.x) {
    atomicAdd(&sums[c], sh[c]);
    atomicAdd(&sums[d + c], sh[d + c]);
  }
}

__global__ void bn_apply(float* __restrict__ x, int n, int d,
                         const float* __restrict__ sums,
                         const float* __restrict__ g, const float* __restrict__ b,
                         int relu)
{
  const int idx = blockIdx.x * blockDim.x + threadIdx.x;
  if (idx >= n * d) return;
  const int c = idx % d;
  const float inv_n = 1.0f / (float)n;
  const float mu  = sums[c] * inv_n;
  const float var = sums[d + c] * inv_n - mu * mu;
  float v = (x[idx] - mu) * rsqrtf(var + 1e-5f) * g[c] + b[c];
  if (relu) v = fmaxf(v, 0.f);
  x[idx] = v;
}

// ---------------------------------------------------------------------------
extern "C" void kernel_launch(void* const* d_in, const int* in_sizes, int n_in,
                              void* d_out, int out_size, void* d_ws, size_t ws_size,
                              hipStream_t stream)
{
  (void)in_sizes; (void)n_in; (void)out_size; (void)ws_size;

  const float* xl    = (const float*)d_in[0];
  const float* xh    = (const float*)d_in[1];
  const float* zstd  = (const float*)d_in[2];
  const int*   si_lh = (const int*)d_in[3];
  const int*   di_lh = (const int*)d_in[4];
  const int*   si_hh = (const int*)d_in[5];
  const int*   di_hh = (const int*)d_in[6];

  const float* d_Wl  = (const float*)d_in[7];
  const float* d_bl  = (const float*)d_in[8];
  const float* d_Wr  = (const float*)d_in[9];
  const float* d_br  = (const float*)d_in[10];
  const float* d_att = (const float*)d_in[11];
  const float* d_b   = (const float*)d_in[12];

  const float* p1_Wl = (const float*)d_in[13];
  const float* p1_bl = (const float*)d_in[14];
  const float* p1_Wr = (const float*)d_in[15];
  const float* p1_br = (const float*)d_in[16];
  const float* p1_at = (const float*)d_in[17];
  const float* p1_b  = (const float*)d_in[18];

  const float* p2_Wl = (const float*)d_in[19];
  const float* p2_bl = (const float*)d_in[20];
  const float* p2_Wr = (const float*)d_in[21];
  const float* p2_br = (const float*)d_in[22];
  const float* p2_at = (const float*)d_in[23];
  const float* p2_b  = (const float*)d_in[24];

  const float* p3_Wl = (const float*)d_in[25];
  const float* p3_bl = (const float*)d_in[26];
  const float* p3_Wr = (const float*)d_in[27];
  const float* p3_br = (const float*)d_in[28];
  const float* p3_at = (const float*)d_in[29];
  const float* p3_b  = (const float*)d_in[30];

  const float* bn0_g = (const float*)d_in[31];
  const float* bn0_b = (const float*)d_in[32];
  const float* bn1_g = (const float*)d_in[33];
  const float* bn1_b = (const float*)d_in[34];
  const float* bn2_g = (const float*)d_in[35];
  const float* bn2_b = (const float*)d_in[36];
  const float* bn3_g = (const float*)d_in[37];
  const float* bn3_b = (const float*)d_in[38];

  const float* prW1  = (const float*)d_in[39];
  const float* prb1  = (const float*)d_in[40];
  const float* prW2  = (const float*)d_in[41];
  const float* prb2  = (const float*)d_in[42];

  // ---- workspace layout (256B aligned) ------------------------------------
  char* ws = (char*)d_ws;
  size_t off = 0;
  auto walloc = [&](size_t bytes) -> void* {
    void* p = ws + off;
    off += (bytes + 255) & ~(size_t)255;
    return p;
  };
  float*    HL  = (float*)walloc((size_t)N_HIGH * 256 * sizeof(float));
  float*    HR  = (float*)walloc((size_t)N_HIGH * 256 * sizeof(float));
  float*    X   = (float*)walloc((size_t)N_HIGH * 257 * sizeof(float));
  float*    Y   = (float*)walloc((size_t)N_HIGH * 256 * sizeof(float));
  float*    LG  = (float*)walloc((size_t)(E_HH + N_HIGH) * 4 * sizeof(float));
  unsigned* Mx  = (unsigned*)walloc((size_t)N_HIGH * 4 * sizeof(unsigned));
  float*    Sm  = (float*)walloc((size_t)N_HIGH * 4 * sizeof(float));
  float*    CNT = (float*)walloc((size_t)N_HIGH * sizeof(float));
  float*    BNS = (float*)walloc(2 * 257 * sizeof(float));

  const unsigned ORD_NEG_INF = 0x007FFFFFu;    // f2ord(-inf)

  auto gemm = [&](const float* A, const float* W, const float* bias, float* C,
                  int M, int K, int N, int relu) {
    dim3 grd((N + 63) / 64, (M + 15) / 16);
    gemm_f32_wmma<<<grd, 128, 0, stream>>>(A, W, bias, C, M, K, N, relu);
  };

  auto gat_edges = [&](const float* hl, const float* hr, const float* att,
                       const int* si, const int* di, int Er, int nloop,
                       int H, int C, float* out, int Nn) {
    const int Et = Er + nloop;
    const int D  = H * C;
    hipMemsetAsync(Sm, 0, (size_t)Nn * H * sizeof(float), stream);
    hipMemsetAsync(CNT, 0, (size_t)Nn * sizeof(float), stream);
    hipMemsetAsync(out, 0, (size_t)Nn * D * sizeof(float), stream);
    fill_u32<<<(Nn * H + 255) / 256, 256, 0, stream>>>(Mx, ORD_NEG_INF, Nn * H);
    edge_logit_max<<<(Et + 7) / 8, 256, 0, stream>>>(hl, hr, att, si, di,
                                                     Er, nloop, H, C, LG, Mx);
    edge_exp_sum<<<(Et * H + 255) / 256, 256, 0, stream>>>(LG, Mx, di, Er, nloop,
                                                           H, Sm, CNT);
    const int tot = Et * D;
    edge_scatter<<<(tot + 255) / 256, 256, 0, stream>>>(hl, si, di, Er, nloop,
                                                        H, C, LG, Sm, out);
  };

  auto bn = [&](float* x, int n, int d, const float* g, const float* b, int relu) {
    hipMemsetAsync(BNS, 0, (size_t)2 * d * sizeof(float), stream);
    bn_reduce<<<(n + 127) / 128, 256, 2 * d * sizeof(float), stream>>>(x, n, d, BNS);
    bn_apply<<<(n * d + 255) / 256, 256, 0, stream>>>(x, n, d, BNS, g, b, relu);
  };

  // ---- downscaler: bipartite GATv2 (low -> high), H=1 C=256 ---------------
  gemm(xl, d_Wl, d_bl, HL, N_LOW, 125, 256, 0);
  gemm(xh, d_Wr, d_br, HR, N_HIGH, 1, 256, 0);
  gat_edges(HL, HR, d_att, si_lh, di_lh, E_LH, 0, 1, 256, Y, N_HIGH);
  finalize_concat<<<(N_HIGH * 257 + 255) / 256, 256, 0, stream>>>(Y, CNT, d_b, zstd,
                                                                  X, N_HIGH);
  bn(X, N_HIGH, 257, bn0_g, bn0_b, 0);

  // ---- p1: H=4 C=64, edges + self loops -----------------------------------
  gemm(X, p1_Wl, p1_bl, HL, N_HIGH, 257, 256, 0);
  gemm(X, p1_Wr, p1_br, HR, N_HIGH, 257, 256, 0);
  gat_edges(HL, HR, p1_at, si_hh, di_hh, E_HH, N_HIGH, 4, 64, Y, N_HIGH);
  finalize_mean<<<(N_HIGH * 256 + 255) / 256, 256, 0, stream>>>(Y, CNT, p1_b, X,
                                                                N_HIGH, 256);
  bn(X, N_HIGH, 256, bn1_g, bn1_b, 1);

  // ---- p2: H=4 C=64 --------------------------------------------------------
  gemm(X, p2_Wl, p2_bl, HL, N_HIGH, 256, 256, 0);
  gemm(X, p2_Wr, p2_br, HR, N_HIGH, 256, 256, 0);
  gat_edges(HL, HR, p2_at, si_hh, di_hh, E_HH, N_HIGH, 4, 64, Y, N_HIGH);
  finalize_mean<<<(N_HIGH * 256 + 255) / 256, 256, 0, stream>>>(Y, CNT, p2_b, X,
                                                                N_HIGH, 256);
  bn(X, N_HIGH, 256, bn2_g, bn2_b, 1);

  // ---- p3: H=1 C=64 --------------------------------------------------------
  gemm(X, p3_Wl, p3_bl, HL, N_HIGH, 256, 64, 0);
  gemm(X, p3_Wr, p3_br, HR, N_HIGH, 256, 64, 0);
  gat_edges(HL, HR, p3_at, si_hh, di_hh, E_HH, N_HIGH, 1, 64, Y, N_HIGH);
  finalize_mean<<<(N_HIGH * 64 + 255) / 256, 256, 0, stream>>>(Y, CNT, p3_b, X,
                                                               N_HIGH, 64);
  bn(X, N_HIGH, 64, bn3_g, bn3_b, 1);

  // ---- predictor MLP -------------------------------------------------------
  gemm(X, prW1, prb1, HL, N_HIGH, 64, 64, 1);
  gemm(HL, prW2, prb2, (float*)d_out, N_HIGH, 64, 1, 0);
}